// GCNConv_56908316672624
// MI455X (gfx1250) — compile-verified
//
#include <hip/hip_runtime.h>
#include <hip/hip_bf16.h>
#include <stdint.h>

typedef __attribute__((ext_vector_type(2))) float v2f;
typedef __attribute__((ext_vector_type(4))) float v4f;
typedef __attribute__((ext_vector_type(8))) float v8f;

#define DIN  128
#define DOUT 128

// ---------------------------------------------------------------------------
// deg[i] = 1.0 (self-loop weight, PyG fill_value=1.0)
__global__ void init_deg_kernel(float* __restrict__ deg, int N) {
    int i = blockIdx.x * blockDim.x + threadIdx.x;
    if (i < N) deg[i] = 1.0f;
}

// deg[col[e]] += w[e]
__global__ void deg_accum_kernel(const long long* __restrict__ col,
                                 const float* __restrict__ w,
                                 float* __restrict__ deg, int E) {
    int e = blockIdx.x * blockDim.x + threadIdx.x;
    if (e < E) {
        unsafeAtomicAdd(&deg[(int)col[e]], w[e]);
    }
}

// dis[i] = deg>0 ? rsqrt(deg) : 0
__global__ void dis_kernel(const float* __restrict__ deg,
                           float* __restrict__ dis, int N) {
    int i = blockIdx.x * blockDim.x + threadIdx.x;
    if (i < N) {
        float d = deg[i];
        dis[i] = (d > 0.0f) ? __frsqrt_rn(d) : 0.0f;
    }
}

// ---------------------------------------------------------------------------
// H = X @ W  via V_WMMA_F32_16X16X4_F32.  One wave -> one 16x16 tile of H.
// Block = 256 threads = 8 waves; wave w covers columns [w*16, w*16+16).
// Grid.x tiles M in steps of 16.
__global__ __launch_bounds__(256)
void gemm_wmma_f32(const float* __restrict__ X, const float* __restrict__ W,
                   float* __restrict__ H, int N) {
    const int lane = threadIdx.x & 31;
    const int wave = threadIdx.x >> 5;
    const int m0   = blockIdx.x * 16;
    const int n0   = wave * 16;
    const int half = lane >> 4;   // 0: lanes 0-15, 1: lanes 16-31
    const int l    = lane & 15;

    // A frag: lane (half,l) holds X[m0+l][k0 + 2*half + {0,1}]
    int arow = m0 + l;
    if (arow >= N) arow = N - 1;            // clamp (no divergence; EXEC stays all-1)
    const float* xrow = X + (size_t)arow * DIN + 2 * half;
    // B frag: VGPR v holds W[k0 + 2*half + v][n0 + l]
    const float* wcol = W + (size_t)(2 * half) * DOUT + n0 + l;

    v8f c = {};
    #pragma unroll 4
    for (int k0 = 0; k0 < DIN; k0 += 4) {
        v2f a = *(const v2f*)(xrow + k0);
        v2f b;
        b.x = wcol[(size_t)k0 * DOUT];
        b.y = wcol[(size_t)(k0 + 1) * DOUT];
        // (neg_a, A, neg_b, B, c_mod, C, reuse_a, reuse_b)
        c = __builtin_amdgcn_wmma_f32_16x16x4_f32(false, a, false, b,
                                                  (short)0, c, false, false);
    }

    // C/D layout: VGPR j -> row m0 + j + 8*half, col n0 + l
    const int rbase = m0 + 8 * half;
    float* hp = H + (size_t)rbase * DOUT + n0 + l;
    if (m0 + 16 <= N) {
        // wave-uniform full-tile fast path: 8 unguarded stores
        #pragma unroll
        for (int j = 0; j < 8; ++j) hp[(size_t)j * DOUT] = c[j];
    } else {
        #pragma unroll
        for (int j = 0; j < 8; ++j) {
            if (rbase + j < N) hp[(size_t)j * DOUT] = c[j];
        }
    }
}

// ---------------------------------------------------------------------------
// out[i][d] = b[d] + H[i][d] * dis[i]^2   (bias + self-loop term; fully
// initializes d_out before the scatter kernel adds edge messages)
__global__ void node_init_kernel(const float* __restrict__ H,
                                 const float* __restrict__ dis,
                                 const float* __restrict__ bias,
                                 float* __restrict__ out, int total) {
    int i = blockIdx.x * blockDim.x + threadIdx.x;
    if (i < total) {
        int node = i >> 7;
        int d    = i & (DOUT - 1);
        float s  = dis[node];
        out[i] = bias[d] + H[i] * (s * s);
    }
}

// One wave per edge: out[col] += H[row] * (dis[row]*w*dis[col]).
// Edge index is wave-uniform -> force scalar (SMEM) loads of row/col/w/dis via
// readfirstlane.  Each lane owns 4 consecutive floats (128/32) -> coalesced
// 512B burst, hardware global_atomic_add_f32 (L2-resident: H and out both fit
// in the 192MB L2, so the random gather/scatter never leaves L2).
__global__ __launch_bounds__(256)
void edge_scatter_kernel(const long long* __restrict__ row,
                         const long long* __restrict__ col,
                         const float* __restrict__ w,
                         const float* __restrict__ dis,
                         const float* __restrict__ H,
                         float* __restrict__ out, int E) {
    int ev   = (int)((blockIdx.x * (size_t)blockDim.x + threadIdx.x) >> 5);
    int e    = __builtin_amdgcn_readfirstlane(ev);   // wave-uniform -> SGPR
    int lane = threadIdx.x & 31;
    if (e >= E) return;

    int r = __builtin_amdgcn_readfirstlane((int)row[e]);
    int c = __builtin_amdgcn_readfirstlane((int)col[e]);
    float norm = dis[r] * w[e] * dis[c];

    v4f m = *((const v4f*)(H + (size_t)r * DOUT) + lane);
    float* op = out + (size_t)c * DOUT + lane * 4;
    unsafeAtomicAdd(op + 0, m.x * norm);
    unsafeAtomicAdd(op + 1, m.y * norm);
    unsafeAtomicAdd(op + 2, m.z * norm);
    unsafeAtomicAdd(op + 3, m.w * norm);
}

// ---------------------------------------------------------------------------
extern "C" void kernel_launch(void* const* d_in, const int* in_sizes, int n_in,
                              void* d_out, int out_size, void* d_ws, size_t ws_size,
                              hipStream_t stream) {
    const float*     x  = (const float*)d_in[0];          // [N,128]
    const long long* ei = (const long long*)d_in[1];      // [2,E] int64
    const float*     ea = (const float*)d_in[2];          // [E]
    const float*     W  = (const float*)d_in[3];          // [128,128]
    const float*     b  = (const float*)d_in[4];          // [128]

    const int N = in_sizes[0] / DIN;
    const int E = in_sizes[2];
    const long long* row = ei;       // edge_index[0]
    const long long* col = ei + E;   // edge_index[1]

    // workspace: deg[N] | dis[N] | H[N*128]
    float* deg = (float*)d_ws;
    float* dis = deg + N;
    float* H   = dis + N;
    float* out = (float*)d_out;

    const int T = 256;

    init_deg_kernel<<<(N + T - 1) / T, T, 0, stream>>>(deg, N);
    deg_accum_kernel<<<(E + T - 1) / T, T, 0, stream>>>(col, ea, deg, E);
    dis_kernel<<<(N + T - 1) / T, T, 0, stream>>>(deg, dis, N);

    const int mTiles = (N + 15) / 16;                   // grid: M tiles; 8 N-tiles/block
    gemm_wmma_f32<<<mTiles, 256, 0, stream>>>(x, W, H, N);

    const int total = N * DOUT;
    node_init_kernel<<<(total + T - 1) / T, T, 0, stream>>>(H, dis, b, out, total);

    const long long threadsE = (long long)E * 32;       // one wave per edge
    edge_scatter_kernel<<<(int)((threadsE + T - 1) / T), T, 0, stream>>>(
        row, col, ea, dis, H, out, E);
}